// EstimateNorm_90185723281989
// MI455X (gfx1250) — compile-verified
//
#include <hip/hip_runtime.h>
#include <hip/hip_bf16.h>
#include <math.h>

typedef float v2f __attribute__((ext_vector_type(2)));
typedef float v8f __attribute__((ext_vector_type(8)));

#define NFACES 256
#define IMG_W  1024
#define IMG_H  1024

// Flat output offsets (float elements), tuple order: xs, IM_composed, imgs_u8, t192, M
#define OFF_XS    ((size_t)0)
#define OFF_IMC   ((size_t)2560)
#define OFF_U8    ((size_t)4096)
#define OFF_T192  ((size_t)38539264)
#define OFF_M     ((size_t)66850816)

__device__ __constant__ float cSRC[5][5][2] = {
  {{103.284f,100.23f},{115.234f,99.98f},{71.48f,138.014f},{102.314f,178.1f},{114.05f,179.404f}},
  {{90.062f,100.236f},{131.136f,101.744f},{79.354f,136.222f},{90.354f,172.38f},{128.492f,173.516f}},
  {{79.46f,102.276f},{144.54f,102.276f},{112.0f,136.986f},{84.926f,174.02f},{139.074f,174.02f}},
  {{93.69f,101.744f},{134.764f,100.236f},{145.474f,136.222f},{96.334f,173.516f},{134.472f,172.38f}},
  {{109.592f,99.98f},{121.542f,100.23f},{153.346f,138.014f},{110.776f,179.404f},{122.514f,178.1f}}
};

__device__ __forceinline__ void invert_affine(const float M[6], float IM[6]) {
  float det = M[0]*M[4] - M[1]*M[3];
  float inv = 1.0f / det;
  IM[0] =  M[4]*inv; IM[1] = -M[1]*inv;
  IM[3] = -M[3]*inv; IM[4] =  M[0]*inv;
  IM[2] = -(IM[0]*M[2] + IM[1]*M[5]);
  IM[5] = -(IM[3]*M[2] + IM[4]*M[5]);
}

// ---------------------------------------------------------------------------
// Stage 1: batched Umeyama fit + argmin over 5 templates; also emits xs copy,
// M, and IM_composed = invert(M) o PREP_INV.  One thread per face.
// ---------------------------------------------------------------------------
__global__ void estimate_norm_kernel(const float* __restrict__ xs, float* out) {
  int f = blockIdx.x * blockDim.x + threadIdx.x;
  if (f >= NFACES) return;

  float kp[5][2];
#pragma unroll
  for (int p = 0; p < 5; ++p) {
    kp[p][0] = xs[(f*5 + p)*2 + 0];
    kp[p][1] = xs[(f*5 + p)*2 + 1];
  }
#pragma unroll
  for (int i = 0; i < 10; ++i) out[OFF_XS + (size_t)f*10 + i] = xs[(size_t)f*10 + i];

  float smx = 0.f, smy = 0.f;
#pragma unroll
  for (int p = 0; p < 5; ++p) { smx += kp[p][0]; smy += kp[p][1]; }
  smx *= 0.2f; smy *= 0.2f;

  float sd[5][2], var_sum = 0.f;
#pragma unroll
  for (int p = 0; p < 5; ++p) {
    sd[p][0] = kp[p][0] - smx; sd[p][1] = kp[p][1] - smy;
    var_sum += sd[p][0]*sd[p][0] + sd[p][1]*sd[p][1];
  }
  var_sum *= 0.2f;

  float bestE = 3.0e38f;
  float bestM[6] = {1.f,0.f,0.f,0.f,1.f,0.f};

  for (int k = 0; k < 5; ++k) {
    float dmx = 0.f, dmy = 0.f;
#pragma unroll
    for (int p = 0; p < 5; ++p) { dmx += cSRC[k][p][0]; dmy += cSRC[k][p][1]; }
    dmx *= 0.2f; dmy *= 0.2f;

    float A00=0.f, A01=0.f, A10=0.f, A11=0.f;
#pragma unroll
    for (int p = 0; p < 5; ++p) {
      float ddx = cSRC[k][p][0] - dmx, ddy = cSRC[k][p][1] - dmy;
      A00 += ddx*sd[p][0]; A01 += ddx*sd[p][1];
      A10 += ddy*sd[p][0]; A11 += ddy*sd[p][1];
    }
    A00 *= 0.2f; A01 *= 0.2f; A10 *= 0.2f; A11 *= 0.2f;

    float detA = A00*A11 - A01*A10;
    float sgn_det = (detA > 0.f) ? 1.f : ((detA < 0.f) ? -1.f : 0.f);

    float E = (A00+A11)*0.5f, F = (A00-A11)*0.5f;
    float G = (A10+A01)*0.5f, H = (A10-A01)*0.5f;
    float Q  = hypotf(E, H), Rr = hypotf(F, G);
    float sxv = Q + Rr, syv = Q - Rr;
    float a1 = atan2f(G, F), a2 = atan2f(H, E);
    float beta = (a2 - a1)*0.5f, gamma = (a2 + a1)*0.5f;
    float cg = cosf(gamma), sg = sinf(gamma);
    float cb = cosf(beta),  sb = sinf(beta);
    float sgn_sy = (syv < 0.f) ? -1.f : 1.f;
    // Vt with second row scaled by sgn_sy
    float Vt00 = cb, Vt01 = -sb, Vt10 = sgn_sy*sb, Vt11 = sgn_sy*cb;
    // R = U @ diag([1, sgn_det]) @ Vt
    float R00 = cg*Vt00 - sg*sgn_det*Vt10;
    float R01 = cg*Vt01 - sg*sgn_det*Vt11;
    float R10 = sg*Vt00 + cg*sgn_det*Vt10;
    float R11 = sg*Vt01 + cg*sgn_det*Vt11;

    float scale = (sxv + fabsf(syv)*sgn_det) / var_sum;
    float tx = dmx - scale*(R00*smx + R01*smy);
    float ty = dmy - scale*(R10*smx + R11*smy);

    float Mk[6] = { scale*R00, scale*R01, tx, scale*R10, scale*R11, ty };

    float e = 0.f;
#pragma unroll
    for (int p = 0; p < 5; ++p) {
      float rx = Mk[0]*kp[p][0] + Mk[1]*kp[p][1] + Mk[2];
      float ry = Mk[3]*kp[p][0] + Mk[4]*kp[p][1] + Mk[5];
      float dx = rx - cSRC[k][p][0], dy = ry - cSRC[k][p][1];
      e += sqrtf(dx*dx + dy*dy);
    }
    if (e < bestE) {
      bestE = e;
#pragma unroll
      for (int i = 0; i < 6; ++i) bestM[i] = Mk[i];
    }
  }

#pragma unroll
  for (int i = 0; i < 6; ++i) out[OFF_M + (size_t)f*6 + i] = bestM[i];

  float IM[6]; invert_affine(bestM, IM);
  // IM_composed = IM o PREP_INV, PREP_INV = [[1.75,0,-56],[0,1.75,-56]]
  out[OFF_IMC + (size_t)f*6 + 0] = IM[0]*1.75f;
  out[OFF_IMC + (size_t)f*6 + 1] = IM[1]*1.75f;
  out[OFF_IMC + (size_t)f*6 + 2] = IM[0]*(-56.f) + IM[1]*(-56.f) + IM[2];
  out[OFF_IMC + (size_t)f*6 + 3] = IM[3]*1.75f;
  out[OFF_IMC + (size_t)f*6 + 4] = IM[4]*1.75f;
  out[OFF_IMC + (size_t)f*6 + 5] = IM[3]*(-56.f) + IM[4]*(-56.f) + IM[5];
}

// ---------------------------------------------------------------------------
// WMMA 16x16 affine grid: D[r][n] = base + a*n + b*r  via V_WMMA_F32_16X16X4_F32
// A(16x4): row r = [1, r, 0, 0]   B(4x16): col n = [base+a*n, b, 0, 0]^T
// Runs on the matrix pipe; co-executes with the VALU gather math.
// ---------------------------------------------------------------------------
__device__ __forceinline__ v8f wmma_affine_grid(float a, float b, float base, int lane) {
  bool lo = lane < 16;
  float col = (float)(lane & 15);
  v2f A, B;
  A.x = lo ? 1.0f : 0.0f;
  A.y = lo ? (float)lane : 0.0f;
  B.x = lo ? (base + a * col) : 0.0f;
  B.y = lo ? b : 0.0f;
  v8f C = {};
  return __builtin_amdgcn_wmma_f32_16x16x4_f32(false, A, false, B, (short)0, C, false, false);
}

// Branch-free bilinear 3-channel gather from HWC 1024x1024x3, zero border:
// clamp addresses (always legal), fold the in-bounds predicate into the tap
// weight.  All 4 b96 loads issue unconditionally and pipeline under LOADcnt.
__device__ __forceinline__ void sample_img3(const float* __restrict__ img,
                                            float px, float py, float o[3]) {
  float xf = floorf(px), yf = floorf(py);
  float fx = px - xf, fy = py - yf;
  int x0 = (int)xf, y0 = (int)yf;
  int x1 = x0 + 1,  y1 = y0 + 1;

  float vx0 = (x0 >= 0 && x0 < IMG_W) ? 1.f : 0.f;
  float vx1 = (x1 >= 0 && x1 < IMG_W) ? 1.f : 0.f;
  float vy0 = (y0 >= 0 && y0 < IMG_H) ? 1.f : 0.f;
  float vy1 = (y1 >= 0 && y1 < IMG_H) ? 1.f : 0.f;

  int cx0 = min(max(x0, 0), IMG_W - 1);
  int cx1 = min(max(x1, 0), IMG_W - 1);
  int cy0 = min(max(y0, 0), IMG_H - 1);
  int cy1 = min(max(y1, 0), IMG_H - 1);

  const float* p00 = img + ((size_t)cy0 * IMG_W + cx0) * 3;
  const float* p01 = img + ((size_t)cy0 * IMG_W + cx1) * 3;
  const float* p10 = img + ((size_t)cy1 * IMG_W + cx0) * 3;
  const float* p11 = img + ((size_t)cy1 * IMG_W + cx1) * 3;

  float w00 = (1.f - fx) * (1.f - fy) * vx0 * vy0;
  float w01 = fx * (1.f - fy) * vx1 * vy0;
  float w10 = (1.f - fx) * fy * vx0 * vy1;
  float w11 = fx * fy * vx1 * vy1;

#pragma unroll
  for (int c = 0; c < 3; ++c)
    o[c] = w00 * p00[c] + w01 * p01[c] + w10 * p10[c] + w11 * p11[c];
}

// ---------------------------------------------------------------------------
// Stage 2: warp to 224x224.  One wave per 16x16 tile (14x14 tiles/face).
// Writes imgs_u8 (as floats, NHWC, non-temporal) and optionally t224 (CHW
// fp32, regular-temporal so the 192-warp's gathers hit L2) into ws.
// ---------------------------------------------------------------------------
__global__ void warp224_kernel(const float* __restrict__ img, float* out, float* t224) {
  int lane = threadIdx.x & 31;
  int tile = blockIdx.x * 8 + (threadIdx.x >> 5);
  int f = tile / 196;
  int t = tile - f * 196;
  int tx0 = (t % 14) * 16;
  int ty0 = (t / 14) * 16;

  float M[6];
#pragma unroll
  for (int i = 0; i < 6; ++i) M[i] = out[OFF_M + (size_t)f*6 + i];
  float IM[6]; invert_affine(M, IM);

  float bx = IM[0]*(float)tx0 + IM[1]*(float)ty0 + IM[2];
  float by = IM[3]*(float)tx0 + IM[4]*(float)ty0 + IM[5];
  v8f SX = wmma_affine_grid(IM[0], IM[1], bx, lane);
  v8f SY = wmma_affine_grid(IM[3], IM[4], by, lane);

  int col = lane & 15;
  int m8 = (lane < 16) ? 0 : 8;
  int ox = tx0 + col;
#pragma unroll
  for (int v = 0; v < 8; ++v) {
    int oy = ty0 + v + m8;
    float o[3]; sample_img3(img, SX[v], SY[v], o);
    size_t u8i = OFF_U8 + (((size_t)f*224 + oy)*224 + ox)*3;
    __builtin_nontemporal_store(floorf(o[0]), out + u8i + 0);
    __builtin_nontemporal_store(floorf(o[1]), out + u8i + 1);
    __builtin_nontemporal_store(floorf(o[2]), out + u8i + 2);
    if (t224) {
      size_t pix = (size_t)oy*224 + ox;
      size_t fb = (size_t)f*3*224*224;
      t224[fb + 0*224*224 + pix] = o[0];
      t224[fb + 1*224*224 + pix] = o[1];
      t224[fb + 2*224*224 + pix] = o[2];
    }
  }
}

// ---------------------------------------------------------------------------
// Stage 3a: warp t224(ws) -> 192x192.  Fixed inverse map (1.75x-56, 1.75y-56).
// ---------------------------------------------------------------------------
__global__ void warp192_ws_kernel(const float* __restrict__ t224, float* out) {
  int lane = threadIdx.x & 31;
  int tile = blockIdx.x * 8 + (threadIdx.x >> 5);
  int f = tile / 144;
  int t = tile - f * 144;
  int tx0 = (t % 12) * 16;
  int ty0 = (t / 12) * 16;

  v8f SX = wmma_affine_grid(1.75f, 0.f, 1.75f*(float)tx0 - 56.f, lane);
  v8f SY = wmma_affine_grid(0.f, 1.75f, 1.75f*(float)ty0 - 56.f, lane);

  int col = lane & 15;
  int m8 = (lane < 16) ? 0 : 8;
  int ox = tx0 + col;
  const float* src = t224 + (size_t)f*3*224*224;
#pragma unroll
  for (int v = 0; v < 8; ++v) {
    int oy = ty0 + v + m8;
    float px = SX[v], py = SY[v];
    float xf = floorf(px), yf = floorf(py);
    float fx = px - xf, fy = py - yf;
    int x0 = (int)xf, y0 = (int)yf;
    int x1 = x0 + 1,  y1 = y0 + 1;

    float vx0 = (x0 >= 0 && x0 < 224) ? 1.f : 0.f;
    float vx1 = (x1 >= 0 && x1 < 224) ? 1.f : 0.f;
    float vy0 = (y0 >= 0 && y0 < 224) ? 1.f : 0.f;
    float vy1 = (y1 >= 0 && y1 < 224) ? 1.f : 0.f;
    int cx0 = min(max(x0, 0), 223), cx1 = min(max(x1, 0), 223);
    int cy0 = min(max(y0, 0), 223), cy1 = min(max(y1, 0), 223);

    float w00 = (1.f - fx) * (1.f - fy) * vx0 * vy0;
    float w01 = fx * (1.f - fy) * vx1 * vy0;
    float w10 = (1.f - fx) * fy * vx0 * vy1;
    float w11 = fx * fy * vx1 * vy1;

    size_t i00 = (size_t)cy0*224 + cx0, i01 = (size_t)cy0*224 + cx1;
    size_t i10 = (size_t)cy1*224 + cx0, i11 = (size_t)cy1*224 + cx1;

    size_t ob = OFF_T192 + (size_t)f*3*192*192 + (size_t)oy*192 + ox;
#pragma unroll
    for (int c = 0; c < 3; ++c) {
      const float* pl = src + (size_t)c*224*224;
      float acc = w00*pl[i00] + w01*pl[i01] + w10*pl[i10] + w11*pl[i11];
      __builtin_nontemporal_store(acc, out + ob + (size_t)c*192*192);
    }
  }
}

// ---------------------------------------------------------------------------
// Stage 3b: same as 3a but recomputes t224 taps from the L2-resident image
// (used if ws is too small to hold t224).
// ---------------------------------------------------------------------------
__global__ void warp192_re_kernel(const float* __restrict__ img, float* out) {
  int lane = threadIdx.x & 31;
  int tile = blockIdx.x * 8 + (threadIdx.x >> 5);
  int f = tile / 144;
  int t = tile - f * 144;
  int tx0 = (t % 12) * 16;
  int ty0 = (t / 12) * 16;

  float M[6];
#pragma unroll
  for (int i = 0; i < 6; ++i) M[i] = out[OFF_M + (size_t)f*6 + i];
  float IM[6]; invert_affine(M, IM);

  v8f SX = wmma_affine_grid(1.75f, 0.f, 1.75f*(float)tx0 - 56.f, lane);
  v8f SY = wmma_affine_grid(0.f, 1.75f, 1.75f*(float)ty0 - 56.f, lane);

  int col = lane & 15;
  int m8 = (lane < 16) ? 0 : 8;
  int ox = tx0 + col;
#pragma unroll
  for (int v = 0; v < 8; ++v) {
    int oy = ty0 + v + m8;
    float px = SX[v], py = SY[v];
    float xf = floorf(px), yf = floorf(py);
    float fx = px - xf, fy = py - yf;
    int x0 = (int)xf, y0 = (int)yf;
    float acc[3] = {0.f, 0.f, 0.f};
#pragma unroll
    for (int dy = 0; dy < 2; ++dy) {
#pragma unroll
      for (int dx = 0; dx < 2; ++dx) {
        int xi = x0 + dx, yi = y0 + dy;
        float valid = (xi >= 0 && xi < 224 && yi >= 0 && yi < 224) ? 1.f : 0.f;
        float w = (dx ? fx : 1.f - fx) * (dy ? fy : 1.f - fy) * valid;
        int cxi = min(max(xi, 0), 223), cyi = min(max(yi, 0), 223);
        float sxx = IM[0]*(float)cxi + IM[1]*(float)cyi + IM[2];
        float syy = IM[3]*(float)cxi + IM[4]*(float)cyi + IM[5];
        float o3[3]; sample_img3(img, sxx, syy, o3);
        acc[0] += w * o3[0]; acc[1] += w * o3[1]; acc[2] += w * o3[2];
      }
    }
    size_t ob = OFF_T192 + (size_t)f*3*192*192 + (size_t)oy*192 + ox;
    __builtin_nontemporal_store(acc[0], out + ob + 0*192*192);
    __builtin_nontemporal_store(acc[1], out + ob + 1*192*192);
    __builtin_nontemporal_store(acc[2], out + ob + 2*192*192);
  }
}

extern "C" void kernel_launch(void* const* d_in, const int* in_sizes, int n_in,
                              void* d_out, int out_size, void* d_ws, size_t ws_size,
                              hipStream_t stream) {
  const float* xs  = (const float*)d_in[0];
  const float* img = (const float*)d_in[1];
  float* out = (float*)d_out;

  estimate_norm_kernel<<<1, 256, 0, stream>>>(xs, out);

  const size_t T224_BYTES = (size_t)NFACES * 3 * 224 * 224 * sizeof(float);
  float* t224 = (ws_size >= T224_BYTES) ? (float*)d_ws : nullptr;

  // 256 faces * 196 tiles / 8 waves-per-block = 6272 blocks, full waves only
  warp224_kernel<<<6272, 256, 0, stream>>>(img, out, t224);

  // 256 faces * 144 tiles / 8 = 4608 blocks
  if (t224) warp192_ws_kernel<<<4608, 256, 0, stream>>>(t224, out);
  else      warp192_re_kernel<<<4608, 256, 0, stream>>>(img, out);
}